// PolicyNetwork_44753559224740
// MI455X (gfx1250) — compile-verified
//
#include <hip/hip_runtime.h>
#include <hip/hip_bf16.h>
#include <math.h>

typedef __attribute__((ext_vector_type(2))) float v2f;
typedef __attribute__((ext_vector_type(8))) float v8f;

#define BSZ   256
#define ASZ   512
#define DDIM  200
#define ADIM  600
#define NTILE 38          // ceil(600/16)
#define X1_STRIDE 608     // padded LDS row stride (keeps float2 alignment)

// ---------------------------------------------------------------------------
// Kernel 1: fused MLP  X2 = relu(obs @ W1 + b1) @ W2 + b2   (f32 WMMA 16x16x4)
// grid = 16 blocks (16 batch rows each), block = 512 threads (16 waves).
// Out-of-range N columns (600..607 in the last tile) are handled by CLAMPING
// the load index instead of predicating: the polluted D columns are simply
// never stored, so the inner loop stays branch-free (EXEC all-1s throughout,
// as WMMA requires) and the compiler can software-pipeline loads + wmma.
// ---------------------------------------------------------------------------
__global__ __launch_bounds__(512) void mlp_wmma_kernel(
    const float* __restrict__ obs, const float* __restrict__ W1,
    const float* __restrict__ b1,  const float* __restrict__ W2,
    const float* __restrict__ b2,  float* __restrict__ X2)
{
  __shared__ float X1[16 * X1_STRIDE];
  const int tid   = threadIdx.x;
  const int wave  = tid >> 5;
  const int lane  = tid & 31;
  const int lhalf = lane >> 4;   // selects K pair (A/B) and M half (C/D)
  const int lmod  = lane & 15;   // M for A, N for B/C/D
  const int m0    = blockIdx.x * 16;

  // ---- GEMM1: X1 = relu(obs[16x200] @ W1[200x600] + b1) ----
  for (int t = wave; t < NTILE; t += 16) {
    const int  n   = t * 16 + lmod;
    const bool nok = (n < ADIM);
    const int  nc  = nok ? n : (ADIM - 1);     // clamp: no branchy loads
    const float bias = b1[nc];
    v8f acc;
#pragma unroll
    for (int v = 0; v < 8; ++v) acc[v] = bias;

    const float* arow = obs + (m0 + lmod) * DDIM + 2 * lhalf;
    const float* bcol = W1 + 2 * lhalf * ADIM + nc;
    for (int k = 0; k < DDIM; k += 4) {        // 50 WMMA steps
      v2f a = *(const v2f*)(arow + k);         // A[16x4]: lane-striped K pair
      v2f bm;
      bm.x = bcol[ k      * ADIM];
      bm.y = bcol[(k + 1) * ADIM];
      acc = __builtin_amdgcn_wmma_f32_16x16x4_f32(false, a, false, bm,
                                                  (short)0, acc, false, false);
    }
    if (nok) {
#pragma unroll
      for (int v = 0; v < 8; ++v)
        X1[(v + 8 * lhalf) * X1_STRIDE + n] = fmaxf(acc[v], 0.0f);
    }
  }
  __syncthreads();

  // ---- GEMM2: X2 = X1[16x600] @ W2[600x600] + b2 ----
  for (int t = wave; t < NTILE; t += 16) {
    const int  n   = t * 16 + lmod;
    const bool nok = (n < ADIM);
    const int  nc  = nok ? n : (ADIM - 1);
    const float bias = b2[nc];
    v8f acc;
#pragma unroll
    for (int v = 0; v < 8; ++v) acc[v] = bias;

    const float* arow = &X1[lmod * X1_STRIDE + 2 * lhalf];
    const float* bcol = W2 + 2 * lhalf * ADIM + nc;
    for (int k = 0; k < ADIM; k += 4) {        // 150 WMMA steps
      v2f a = *(const v2f*)(arow + k);         // LDS float2 read
      v2f bm;
      bm.x = bcol[ k      * ADIM];
      bm.y = bcol[(k + 1) * ADIM];
      acc = __builtin_amdgcn_wmma_f32_16x16x4_f32(false, a, false, bm,
                                                  (short)0, acc, false, false);
    }
    if (nok) {
#pragma unroll
      for (int v = 0; v < 8; ++v)
        X2[(m0 + v + 8 * lhalf) * ADIM + n] = acc[v];
    }
  }
}

// ---------------------------------------------------------------------------
// Kernel 2: gather-dot scoring + softmax + entropy (bandwidth-bound part).
// grid = 256 (one block per batch row), block = 256 threads (8 waves).
// One wave per action: 32 lanes stream each 800-byte embedding row with
// float2 (global_load_b64) coalesced loads; next action's HBM-resident rows
// (ent/trip) are prefetched (global_prefetch_b8) one iteration ahead.
// ---------------------------------------------------------------------------
__global__ __launch_bounds__(256) void score_softmax_kernel(
    const float* __restrict__ X2,
    const float* __restrict__ rel,  const float* __restrict__ entb,
    const float* __restrict__ trip,
    const int* __restrict__ r_sp,   const int* __restrict__ e_sp,
    const int* __restrict__ t_id,   const int* __restrict__ amask,
    float* __restrict__ dist,       float* __restrict__ ent_out)
{
  __shared__ float xv[ADIM];
  __shared__ float logits[ASZ];
  __shared__ float red[256];
  const int b    = blockIdx.x;
  const int tid  = threadIdx.x;
  const int wave = tid >> 5;
  const int lane = tid & 31;

  for (int i = tid; i < ADIM; i += 256) xv[i] = X2[b * ADIM + i];
  __syncthreads();

  for (int a = wave; a < ASZ; a += 8) {
    const int idx = b * ASZ + a;
    const int r = r_sp[idx];
    const int e = e_sp[idx];
    const int t = t_id[idx];
    const float* rp = rel  + (long)r * DDIM;
    const float* ep = entb + (long)e * DDIM;
    const float* tp = trip + (long)t * DDIM;

    // Prefetch next action's HBM-heavy rows (ent: 80MB, trip: 160MB tables).
    const int an = a + 8;
    if (an < ASZ && lane < 7) {               // 7 x 128B lines cover 800B row
      const int idxn = b * ASZ + an;
      const char* epn = (const char*)(entb + (long)e_sp[idxn] * DDIM);
      const char* tpn = (const char*)(trip + (long)t_id[idxn] * DDIM);
      __builtin_prefetch(epn + lane * 128, 0, 0);
      __builtin_prefetch(tpn + lane * 128, 0, 0);
    }

    float s = 0.0f;
    for (int j = lane; j < DDIM / 2; j += 32) {   // float2-vectorized dot
      const v2f rv = *(const v2f*)(rp + 2 * j);
      const v2f ev = *(const v2f*)(ep + 2 * j);
      const v2f tv = *(const v2f*)(tp + 2 * j);
      s += rv.x * xv[2 * j]            + rv.y * xv[2 * j + 1];
      s += ev.x * xv[DDIM + 2 * j]     + ev.y * xv[DDIM + 2 * j + 1];
      s += tv.x * xv[2 * DDIM + 2 * j] + tv.y * xv[2 * DDIM + 2 * j + 1];
    }
#pragma unroll
    for (int o = 16; o >= 1; o >>= 1) s += __shfl_xor(s, o, 32);
    if (lane == 0)
      logits[a] = s - (1.0f - (float)amask[idx]) * 1.0e31f;
  }
  __syncthreads();

  // --- softmax over 512 logits (2 per thread), max-subtracted like jax ---
  const float l0 = logits[tid];
  const float l1 = logits[tid + 256];
  red[tid] = fmaxf(l0, l1);
  __syncthreads();
  for (int sft = 128; sft >= 1; sft >>= 1) {
    if (tid < sft) red[tid] = fmaxf(red[tid], red[tid + sft]);
    __syncthreads();
  }
  const float mx = red[0];
  __syncthreads();

  const float p0 = expf(l0 - mx);
  const float p1 = expf(l1 - mx);
  red[tid] = p0 + p1;
  __syncthreads();
  for (int sft = 128; sft >= 1; sft >>= 1) {
    if (tid < sft) red[tid] += red[tid + sft];
    __syncthreads();
  }
  const float Z = red[0];
  __syncthreads();

  const float inv = 1.0f / Z;
  const float d0 = p0 * inv;
  const float d1 = p1 * inv;
  dist[b * ASZ + tid]       = d0;
  dist[b * ASZ + tid + 256] = d1;

  const float EPS = 2.2204460492503131e-16f;  // float64 eps, as in reference
  red[tid] = -d0 * logf(d0 + EPS) - d1 * logf(d1 + EPS);
  __syncthreads();
  for (int sft = 128; sft >= 1; sft >>= 1) {
    if (tid < sft) red[tid] += red[tid + sft];
    __syncthreads();
  }
  if (tid == 0) ent_out[b] = red[0];
}

// ---------------------------------------------------------------------------
extern "C" void kernel_launch(void* const* d_in, const int* in_sizes, int n_in,
                              void* d_out, int out_size, void* d_ws, size_t ws_size,
                              hipStream_t stream) {
  const float* obs  = (const float*)d_in[0];
  const float* W1   = (const float*)d_in[1];
  const float* b1   = (const float*)d_in[2];
  const float* W2   = (const float*)d_in[3];
  const float* b2   = (const float*)d_in[4];
  const float* rel  = (const float*)d_in[5];
  const float* ente = (const float*)d_in[6];
  const float* trip = (const float*)d_in[7];
  const int*   r_sp = (const int*)d_in[8];
  const int*   e_sp = (const int*)d_in[9];
  const int*   t_id = (const int*)d_in[10];
  const int*   amsk = (const int*)d_in[11];

  float* dist = (float*)d_out;            // [256*512] action_dist
  float* entv = dist + BSZ * ASZ;         // [256] entropy
  float* X2   = (float*)d_ws;             // scratch: 256*600 f32 = 600 KB

  mlp_wmma_kernel<<<BSZ / 16, 512, 0, stream>>>(obs, W1, b1, W2, b2, X2);
  score_softmax_kernel<<<BSZ, 256, 0, stream>>>(X2, rel, ente, trip,
                                                r_sp, e_sp, t_id, amsk,
                                                dist, entv);
}